// GATEncoder_19859928777441
// MI455X (gfx1250) — compile-verified
//
#include <hip/hip_runtime.h>
#include <hip/hip_bf16.h>
#include <math.h>

typedef __attribute__((ext_vector_type(2))) float v2f;
typedef __attribute__((ext_vector_type(8))) float v8f;

// ---------------------------------------------------------------- utilities
__global__ void k_fill(float* __restrict__ p, float v, int n) {
    int i = blockIdx.x * blockDim.x + threadIdx.x;
    if (i < n) p[i] = v;
}

__global__ void k_transpose(const float* __restrict__ W, float* __restrict__ Wt,
                            int K, int Ncol) {
    int i = blockIdx.x * blockDim.x + threadIdx.x;
    if (i < K * Ncol) {
        int k = i / Ncol, nn = i % Ncol;
        Wt[nn * K + k] = W[i];
    }
}

// ---------------------------------------------------------------- f32 WMMA GEMM
// C[nRows x nCols] = A[nRows x 128] * W[128 x nCols], with W pre-transposed as
// Wt[nCols x 128].  One wave per 16x16 tile; blockDim=(32,4): 4 waves share the
// same 16-row A stripe (L1 reuse).  nRows multiple of 16 (50000 = 3125*16),
// nCols multiple of 64.  No divergence -> EXEC all-ones for WMMA.
__global__ void k_gemm_f32_wmma(const float* __restrict__ A,
                                const float* __restrict__ Wt,
                                float* __restrict__ C, int nCols) {
    const int K = 128;
    int lane = threadIdx.x;          // 0..31
    int half = lane >> 4;            // 0: K pair {0,1}, 1: K pair {2,3}
    int l    = lane & 15;
    int rowBase = blockIdx.x * 16;
    int colBase = (blockIdx.y * 4 + threadIdx.y) * 16;

    const float* arow = A  + (size_t)(rowBase + l) * K + 2 * half;
    const float* brow = Wt + (size_t)(colBase + l) * K + 2 * half;

    v8f acc = {};
#pragma unroll
    for (int k = 0; k < K; k += 4) {
        v2f a = *(const v2f*)(arow + k);   // A[m][k+2h], A[m][k+2h+1]
        v2f b = *(const v2f*)(brow + k);   // B[k+2h][n], B[k+2h+1][n]
        acc = __builtin_amdgcn_wmma_f32_16x16x4_f32(
            /*neg_a=*/false, a, /*neg_b=*/false, b,
            /*c_mod=*/(short)0, acc, /*reuse_a=*/false, /*reuse_b=*/false);
    }
    // C/D layout: VGPR r -> (M = r + 8*half, N = l)
#pragma unroll
    for (int r = 0; r < 8; ++r)
        C[(size_t)(rowBase + r + 8 * half) * nCols + colBase + l] = acc[r];
}

// ---------------------------------------------------------------- attention scores
// asrc[node*H+h] = sum_c h[node,h,c]*att_src[h,c]   (same for adst)
__global__ void k_attn(const float* __restrict__ h, const float* __restrict__ att_s,
                       const float* __restrict__ att_d, float* __restrict__ asrc,
                       float* __restrict__ adst, int n, int heads, int C) {
    int i = blockIdx.x * blockDim.x + threadIdx.x;
    if (i >= n * heads) return;
    int node = i / heads, hd = i % heads;
    const float* hp = h + (size_t)node * heads * C + (size_t)hd * C;
    float ss = 0.f, sd = 0.f;
    for (int c = 0; c < C; ++c) {
        float v = hp[c];
        ss += v * att_s[hd * C + c];
        sd += v * att_d[hd * C + c];
    }
    asrc[i] = ss;
    adst[i] = sd;
}

// ---------------------------------------------------------------- edge passes
__device__ __forceinline__ void atomicMaxF(float* addr, float val) {
    // ordered-float trick; correct with -inf init (0xFF800000)
    if (val >= 0.f) atomicMax((int*)addr, __float_as_int(val));
    else            atomicMin((unsigned int*)addr, (unsigned int)__float_as_int(val));
}

__device__ __forceinline__ void edge_pair(int e, int E, const int* __restrict__ src,
                                          const int* __restrict__ dst, int& s, int& d) {
    if (e < E) { s = src[e]; d = dst[e]; }
    else       { s = d = e - E; }        // appended self loops
}

__global__ void k_edge_max(const int* __restrict__ src, const int* __restrict__ dst,
                           const float* __restrict__ asrc, const float* __restrict__ adst,
                           float* __restrict__ m, int E, int n, int heads) {
    int e = blockIdx.x * blockDim.x + threadIdx.x;
    if (e >= E + n) return;
    int s, d; edge_pair(e, E, src, dst, s, d);
    for (int h = 0; h < heads; ++h) {
        float v = asrc[s * heads + h] + adst[d * heads + h];
        v = v > 0.f ? v : 0.2f * v;                 // leaky_relu(0.2)
        atomicMaxF(&m[d * heads + h], v);
    }
}

__global__ void k_edge_expsum(const int* __restrict__ src, const int* __restrict__ dst,
                              const float* __restrict__ asrc, const float* __restrict__ adst,
                              const float* __restrict__ m, float* __restrict__ ex,
                              float* __restrict__ den, int E, int n, int heads) {
    int e = blockIdx.x * blockDim.x + threadIdx.x;
    if (e >= E + n) return;
    int s, d; edge_pair(e, E, src, dst, s, d);
    for (int h = 0; h < heads; ++h) {
        float v = asrc[s * heads + h] + adst[d * heads + h];
        v = v > 0.f ? v : 0.2f * v;
        float x = expf(v - m[d * heads + h]);
        ex[(size_t)e * heads + h] = x;
        atomicAdd(&den[d * heads + h], x);
    }
}

// one edge per (HC) contiguous threads: coalesced gather of feat[src], L2-local
// atomic scatter into agg[dst]
__global__ void k_edge_agg(const int* __restrict__ src, const int* __restrict__ dst,
                           const float* __restrict__ ex, const float* __restrict__ den,
                           const float* __restrict__ feat, float* __restrict__ agg,
                           int E, int n, int heads, int C) {
    int HC = heads * C;
    int epb = blockDim.x / HC;
    int eLocal = threadIdx.x / HC;
    int c = threadIdx.x % HC;
    int e = blockIdx.x * epb + eLocal;
    if (e >= E + n) return;
    int s, d; edge_pair(e, E, src, dst, s, d);
    int hd = c / C;
    float alpha = ex[(size_t)e * heads + hd] / den[d * heads + hd];
    atomicAdd(&agg[(size_t)d * HC + c], feat[(size_t)s * HC + c] * alpha);
}

// ---------------------------------------------------------------- epilogues
__global__ void k_bias_elu(float* __restrict__ h, const float* __restrict__ bias,
                           int n, int HC) {
    int i = blockIdx.x * blockDim.x + threadIdx.x;
    if (i >= n * HC) return;
    float v = h[i] + bias[i % HC];
    h[i] = v > 0.f ? v : (expf(v) - 1.f);
}

__global__ void k_bias_add(float* __restrict__ o, const float* __restrict__ bias,
                           int n, int C) {
    int i = blockIdx.x * blockDim.x + threadIdx.x;
    if (i >= n * C) return;
    o[i] += bias[i % C];
}

// ---------------------------------------------------------------- launcher
extern "C" void kernel_launch(void* const* d_in, const int* in_sizes, int n_in,
                              void* d_out, int out_size, void* d_ws, size_t ws_size,
                              hipStream_t stream) {
    const float* x      = (const float*)d_in[0];
    const int*   ei     = (const int*)  d_in[1];
    const float* W1     = (const float*)d_in[2];
    const float* att_s1 = (const float*)d_in[3];
    const float* att_d1 = (const float*)d_in[4];
    const float* b1     = (const float*)d_in[5];
    const float* W2     = (const float*)d_in[6];
    const float* att_s2 = (const float*)d_in[7];
    const float* att_d2 = (const float*)d_in[8];
    const float* b2     = (const float*)d_in[9];

    const int IN = 128, HID = 128, H1 = 4, C1 = 32, OUT = 64;
    const int n   = in_sizes[0] / IN;   // 50000
    const int E   = in_sizes[1] / 2;    // 800000
    const int tot = E + n;              // edges + self loops

    // d_out = [out (n*64)] ++ [h (n*128)]
    float* out1  = (float*)d_out;
    float* hfeat = (float*)d_out + (size_t)n * OUT;

    float* w = (float*)d_ws;
    size_t off = 0;
    float* Wt1 = w + off; off += (size_t)IN * HID;
    float* Wt2 = w + off; off += (size_t)HID * OUT;
    float* h1  = w + off; off += (size_t)n * HID;
    float* h2  = w + off; off += (size_t)n * OUT;
    float* as1 = w + off; off += (size_t)n * H1;
    float* ad1 = w + off; off += (size_t)n * H1;
    float* m1  = w + off; off += (size_t)n * H1;
    float* dn1 = w + off; off += (size_t)n * H1;
    float* as2 = w + off; off += (size_t)n;
    float* ad2 = w + off; off += (size_t)n;
    float* m2  = w + off; off += (size_t)n;
    float* dn2 = w + off; off += (size_t)n;
    float* ex1 = w + off; off += (size_t)tot * H1;
    float* ex2 = w + off; off += (size_t)tot;

    const int* srcI = ei;
    const int* dstI = ei + E;

    auto cdiv = [](long a, long b) { return (unsigned)((a + b - 1) / b); };
    dim3 B256(256);

    k_transpose<<<cdiv((long)IN * HID, 256), B256, 0, stream>>>(W1, Wt1, IN, HID);
    k_transpose<<<cdiv((long)HID * OUT, 256), B256, 0, stream>>>(W2, Wt2, HID, OUT);

    // ---------------- layer 1 ----------------
    k_gemm_f32_wmma<<<dim3(n / 16, HID / 64), dim3(32, 4), 0, stream>>>(x, Wt1, h1, HID);
    k_attn<<<cdiv((long)n * H1, 256), B256, 0, stream>>>(h1, att_s1, att_d1, as1, ad1, n, H1, C1);

    k_fill<<<cdiv((long)n * H1, 256), B256, 0, stream>>>(m1, -INFINITY, n * H1);
    k_fill<<<cdiv((long)n * H1, 256), B256, 0, stream>>>(dn1, 0.f, n * H1);
    k_fill<<<cdiv((long)n * HID, 256), B256, 0, stream>>>(hfeat, 0.f, n * HID);

    k_edge_max   <<<cdiv(tot, 256), B256, 0, stream>>>(srcI, dstI, as1, ad1, m1, E, n, H1);
    k_edge_expsum<<<cdiv(tot, 256), B256, 0, stream>>>(srcI, dstI, as1, ad1, m1, ex1, dn1, E, n, H1);
    k_edge_agg   <<<cdiv((long)tot * 128, 256), B256, 0, stream>>>(srcI, dstI, ex1, dn1, h1, hfeat, E, n, H1, C1);
    k_bias_elu   <<<cdiv((long)n * HID, 256), B256, 0, stream>>>(hfeat, b1, n, HID);

    // ---------------- layer 2 ----------------
    k_gemm_f32_wmma<<<dim3(n / 16, OUT / 64), dim3(32, 4), 0, stream>>>(hfeat, Wt2, h2, OUT);
    k_attn<<<cdiv((long)n, 256), B256, 0, stream>>>(h2, att_s2, att_d2, as2, ad2, n, 1, OUT);

    k_fill<<<cdiv((long)n, 256), B256, 0, stream>>>(m2, -INFINITY, n);
    k_fill<<<cdiv((long)n, 256), B256, 0, stream>>>(dn2, 0.f, n);
    k_fill<<<cdiv((long)n * OUT, 256), B256, 0, stream>>>(out1, 0.f, n * OUT);

    k_edge_max   <<<cdiv(tot, 256), B256, 0, stream>>>(srcI, dstI, as2, ad2, m2, E, n, 1);
    k_edge_expsum<<<cdiv(tot, 256), B256, 0, stream>>>(srcI, dstI, as2, ad2, m2, ex2, dn2, E, n, 1);
    k_edge_agg   <<<cdiv((long)tot * 64, 256), B256, 0, stream>>>(srcI, dstI, ex2, dn2, h2, out1, E, n, 1, OUT);
    k_bias_add   <<<cdiv((long)n * OUT, 256), B256, 0, stream>>>(out1, b2, n, OUT);
}